// Blocks_71803263254757
// MI455X (gfx1250) — compile-verified
//
#include <hip/hip_runtime.h>
#include <hip/hip_bf16.h>

#define Bdim 8
#define Ldim 16384
#define Hdim 64
#define NST  32             // N_HALF modes
#define NSEG 64
#define SEG  (Ldim / NSEG)  // 256
#define Tc   16             // chunk length (WMMA M/K)
#define CHPSEG (SEG / Tc)   // 16 chunks per segment
#define NSQ  (Bdim * NSEG)  // 512 independent sequences per channel

typedef __attribute__((ext_vector_type(2))) float v2f;
typedef __attribute__((ext_vector_type(8))) float v8f;

// ---------------------------------------------------------------------------
// Kernel A: per-(h,n) w, 2*C2 (sign-folded) and FiLM gains. Tiny.
// wc2[h*32+n] = {wr, wi, 2*C2r, -2*C2i};  gb[b*2H+j]
// ---------------------------------------------------------------------------
__global__ void precompute_kernel(const float* __restrict__ log_dt,
                                  const float* __restrict__ log_A_real,
                                  const float* __restrict__ A_imag,
                                  const float* __restrict__ C_ri,
                                  const float* __restrict__ cond,
                                  const float* __restrict__ W_film,
                                  const float* __restrict__ b_film,
                                  float* __restrict__ wc2,
                                  float* __restrict__ gb) {
  int t = blockIdx.x * blockDim.x + threadIdx.x;
  if (t < Hdim * NST) {
    int h = t >> 5;
    float dt = __expf(log_dt[h]);
    float Ar = -__expf(log_A_real[t]);
    float Ai = A_imag[t];
    float dr = Ar * dt, di = Ai * dt;
    float ew = __expf(dr);
    float wr = ew * __cosf(di);
    float wi = ew * __sinf(di);
    float nr = wr - 1.f, ni = wi;              // q = (w-1)/A
    float inv = 1.f / (Ar * Ar + Ai * Ai);
    float qr = (nr * Ar + ni * Ai) * inv;
    float qi = (ni * Ar - nr * Ai) * inv;
    float Cr = C_ri[2 * t], Ci = C_ri[2 * t + 1];
    float c2r = Cr * qr - Ci * qi;
    float c2i = Cr * qi + Ci * qr;
    ((float4*)wc2)[t] = make_float4(wr, wi, 2.f * c2r, -2.f * c2i);
  }
  if (t < Bdim * 2 * Hdim) {
    int b = t / (2 * Hdim);
    int j = t - b * (2 * Hdim);
    gb[t] = b_film[j] + cond[b * 2 + 0] * W_film[j * 2 + 0]
                      + cond[b * 2 + 1] * W_film[j * 2 + 1];
  }
}

// ---------------------------------------------------------------------------
// Kernel B: build the per-h chunk matrices.
//  Phi[h][t][2n]=2Re(C2 w^{t+1}), [2n+1]=-2Im(C2 w^{t+1})   (16x64, A-matrix)
//  Psi[h][2n][tau]=Re(w^{15-tau}), [2n+1][tau]=Im(w^{15-tau}) (64x16, A-matrix)
//  lam2[h][n]=w^16,  wSb[h][n]=w^SEG,  kh[h][t]=2Re(Sum C2 w^t)(+D at t=0)
// ---------------------------------------------------------------------------
__global__ void precompute2_kernel(const float* __restrict__ log_dt,
                                   const float* __restrict__ log_A_real,
                                   const float* __restrict__ A_imag,
                                   const float* __restrict__ Dv,
                                   const float* __restrict__ wc2,
                                   float* __restrict__ Phi,
                                   float* __restrict__ Psi,
                                   float* __restrict__ lam2,
                                   float* __restrict__ wSb,
                                   float* __restrict__ kh) {
  int t = blockIdx.x * blockDim.x + threadIdx.x;   // H*NST*Tc = 32768
  int tt = t & 15;
  int n  = (t >> 4) & 31;
  int h  = t >> 9;
  float dt = __expf(log_dt[h]);
  float Ar = -__expf(log_A_real[h * NST + n]);
  float Ai = A_imag[h * NST + n];
  float dr = Ar * dt, di = Ai * dt;
  float4 c = ((const float4*)wc2)[h * NST + n];    // (wr,wi, 2C2r, -2C2i)

  {  // Phi row tt, cols 2n / 2n+1 : w^(tt+1)
    float p = (float)(tt + 1);
    float e = __expf(dr * p);
    float pr = e * __cosf(di * p), pi = e * __sinf(di * p);
    float* row = Phi + ((long)h * Tc + tt) * 64;
    row[2 * n]     =  c.z * pr + c.w * pi;   // 2Re(C2 w^p)
    row[2 * n + 1] = -c.z * pi + c.w * pr;   // -2Im(C2 w^p)
  }
  {  // Psi rows 2n/2n+1, col tt : w^(15-tt)
    float p = (float)(15 - tt);
    float e = __expf(dr * p);
    float qr = e * __cosf(di * p), qi = e * __sinf(di * p);
    Psi[((long)h * 64 + 2 * n) * Tc + tt]     = qr;
    Psi[((long)h * 64 + 2 * n + 1) * Tc + tt] = qi;
  }
  if (tt == 0) {  // lambda = w^16
    float e = __expf(dr * 16.f);
    ((float2*)lam2)[h * NST + n] = make_float2(e * __cosf(di * 16.f),
                                               e * __sinf(di * 16.f));
  }
  if (tt == 1) {  // w^SEG
    float e = __expf(dr * (float)SEG);
    ((float2*)wSb)[h * NST + n] = make_float2(e * __cosf(di * (float)SEG),
                                              e * __sinf(di * (float)SEG));
  }
  if (t < Hdim * Tc) {  // kh role: (hh, th)
    int th = t & 15, hh = t >> 4;
    float dth = __expf(log_dt[hh]);
    float sum = (th == 0) ? Dv[hh] : 0.f;
    for (int m = 0; m < NST; ++m) {
      float ar = -__expf(log_A_real[hh * NST + m]);
      float ai = A_imag[hh * NST + m];
      float e = __expf(ar * dth * (float)th);
      float pr = e * __cosf(ai * dth * (float)th);
      float pi = e * __sinf(ai * dth * (float)th);
      float4 cc = ((const float4*)wc2)[hh * NST + m];
      sum += cc.z * pr + cc.w * pi;
    }
    kh[hh * Tc + th] = sum;
  }
}

// ---------------------------------------------------------------------------
// Kernel C: h = relu(x @ W_lin^T + b_lin) via fp32 WMMA, stored transposed as
// u[b, h, l]. One wave per 16x16 output tile.
// ---------------------------------------------------------------------------
__global__ void gemm_relu_tr_kernel(const float* __restrict__ x,
                                    const float* __restrict__ W,
                                    const float* __restrict__ bias,
                                    float* __restrict__ u) {
  int wave = (blockIdx.x * blockDim.x + threadIdx.x) >> 5;
  int lane = threadIdx.x & 31;
  int col  = wave & 3;
  long row0 = (long)(wave >> 2) * 16;
  int j0 = col * 16;
  int half = lane >> 4;
  int lid  = lane & 15;

  const float* xrow = x + (row0 + lid) * Hdim + half * 2;
  const float* wrow = W + (long)(j0 + lid) * Hdim + half * 2;

  v8f acc = {0.f, 0.f, 0.f, 0.f, 0.f, 0.f, 0.f, 0.f};
#pragma unroll
  for (int k0 = 0; k0 < Hdim; k0 += 4) {
    v2f a  = *(const v2f*)(xrow + k0);
    v2f bf = *(const v2f*)(wrow + k0);
    acc = __builtin_amdgcn_wmma_f32_16x16x4_f32(
        false, a, false, bf, (short)0, acc, false, false);
  }
  float bj = bias[j0 + lid];
#pragma unroll
  for (int r = 0; r < 8; ++r) {
    float v = acc[r] + bj;
    acc[r] = v > 0.f ? v : 0.f;
  }
  long b  = row0 / Ldim;
  long l0 = row0 & (Ldim - 1);
  float* dst = u + (b * Hdim + j0 + lid) * (long)Ldim + l0 + half * 8;
#pragma unroll
  for (int r = 0; r < 8; ++r) dst[r] = acc[r];
}

// ---------------------------------------------------------------------------
// Kernel D: chunked SSM conv via WMMA. Wave = (h, tile of 16 sequences);
// sequences are (b,seg) pairs with zero initial state. 16 chunks of 16 steps.
//   Y      = TrilK @ U + Phi0 @ S        (20 wmma, D-layout, stored in place)
//   dS     = Psi @ U                     (16 wmma, D-layout)
//   S'     = Lam (.) S + relayout(dS)    (lane-pair shfl_xor(16) relayout)
// End state written to Ebuf.
// ---------------------------------------------------------------------------
__global__ void conv_scan_kernel(float* __restrict__ uy,
                                 const float* __restrict__ kh,
                                 const float* __restrict__ Phi,
                                 const float* __restrict__ Psi,
                                 const float* __restrict__ lam2,
                                 float* __restrict__ Ebuf) {
  int wave = (blockIdx.x * blockDim.x + threadIdx.x) >> 5;  // h*32 + tile
  int lane = threadIdx.x & 31;
  int tile = wave & 31;
  int h    = wave >> 5;
  int lid  = lane & 15;
  int bit  = lane >> 4;

  int seq  = tile * 16 + lid;          // this lane's column
  int bcol = seq >> 6;                 // /NSEG
  int segc = seq & (NSEG - 1);
  float* colbase = uy + ((long)(bcol * Hdim + h)) * Ldim + segc * SEG;

  // A fragments (chunk-invariant)
  v2f tk[4];
  const float* khh = kh + h * Tc;
#pragma unroll
  for (int ks = 0; ks < 4; ++ks) {
    int k0 = 4 * ks + 2 * bit;
    float a0 = (lid >= k0)     ? khh[lid - k0]     : 0.f;
    float a1 = (lid >= k0 + 1) ? khh[lid - k0 - 1] : 0.f;
    v2f tv = {a0, a1};
    tk[ks] = tv;
  }
  v2f ph[16];
  const float* phirow = Phi + ((long)h * Tc + lid) * 64 + 2 * bit;
#pragma unroll
  for (int ks = 0; ks < 16; ++ks) ph[ks] = *(const v2f*)(phirow + 4 * ks);
  v2f psf[4][4];
#pragma unroll
  for (int mt = 0; mt < 4; ++mt) {
    const float* prow = Psi + ((long)h * 64 + 16 * mt + lid) * Tc + 2 * bit;
#pragma unroll
    for (int ks = 0; ks < 4; ++ks) psf[mt][ks] = *(const v2f*)(prow + 4 * ks);
  }
  float2 lm[16];
  const float2* lamh = ((const float2*)lam2) + h * NST + bit;
#pragma unroll
  for (int m = 0; m < 16; ++m) lm[m] = lamh[2 * m];

  v2f SB[16];
#pragma unroll
  for (int m = 0; m < 16; ++m) { v2f z = {0.f, 0.f}; SB[m] = z; }

  for (int c = 0; c < CHPSEG; ++c) {
    float* cb = colbase + c * Tc;
    v2f ub[4];
#pragma unroll
    for (int ks = 0; ks < 4; ++ks)
      ub[ks] = *(const v2f*)(cb + 4 * ks + 2 * bit);

    v8f acc = {0.f, 0.f, 0.f, 0.f, 0.f, 0.f, 0.f, 0.f};
#pragma unroll
    for (int ks = 0; ks < 4; ++ks)
      acc = __builtin_amdgcn_wmma_f32_16x16x4_f32(
          false, tk[ks], false, ub[ks], (short)0, acc, false, false);
#pragma unroll
    for (int ks = 0; ks < 16; ++ks)
      acc = __builtin_amdgcn_wmma_f32_16x16x4_f32(
          false, ph[ks], false, SB[ks], (short)0, acc, false, false);

    v8f ds[4];
#pragma unroll
    for (int mt = 0; mt < 4; ++mt) {
      v8f a2 = {0.f, 0.f, 0.f, 0.f, 0.f, 0.f, 0.f, 0.f};
#pragma unroll
      for (int ks = 0; ks < 4; ++ks)
        a2 = __builtin_amdgcn_wmma_f32_16x16x4_f32(
            false, psf[mt][ks], false, ub[ks], (short)0, a2, false, false);
      ds[mt] = a2;
    }

    // store Y (in place): lane holds times c*16 + 8*bit + r of its column
    float* yb = cb + 8 * bit;
#pragma unroll
    for (int r = 0; r < 8; ++r) yb[r] = acc[r];

    // S' = Lam (.) S + dS  (D-layout -> B-fragment layout via lane-pair xchg)
#pragma unroll
    for (int mt = 0; mt < 4; ++mt) {
      float oth[8];
#pragma unroll
      for (int r = 0; r < 8; ++r) oth[r] = __shfl_xor(ds[mt][r], 16, 32);
#pragma unroll
      for (int j = 0; j < 4; ++j) {
        const int kt0 = 4 * j;                    // bit==0: Re k_local
        const int kt1 = 4 * j + 2;                // bit==1: Re k_local
        const int r0 = kt0 & 7; const bool own0 = ((kt0 >> 3) == 0);
        const int r1 = kt1 & 7; const bool own1 = ((kt1 >> 3) == 1);
        float vr0 = own0 ? ds[mt][r0]     : oth[r0];
        float vi0 = own0 ? ds[mt][r0 + 1] : oth[r0 + 1];
        float vr1 = own1 ? ds[mt][r1]     : oth[r1];
        float vi1 = own1 ? ds[mt][r1 + 1] : oth[r1 + 1];
        float vr = bit ? vr1 : vr0;
        float vi = bit ? vi1 : vi0;
        const int m = 4 * mt + j;
        float sr = SB[m][0], si = SB[m][1];
        float nr = fmaf(lm[m].x, sr, fmaf(-lm[m].y, si, vr));
        float ni = fmaf(lm[m].x, si, fmaf(lm[m].y, sr, vi));
        SB[m][0] = nr; SB[m][1] = ni;
      }
    }
  }
  // end-of-segment states
  float2* eb = ((float2*)Ebuf) + ((long)h * NSQ + seq) * NST;
#pragma unroll
  for (int m = 0; m < 16; ++m)
    eb[2 * m + bit] = make_float2(SB[m][0], SB[m][1]);
}

// ---------------------------------------------------------------------------
// Kernel E: per-(b,h) wave, lane = mode: serial prefix over NSEG end-states.
// ---------------------------------------------------------------------------
__global__ void scan_combine_kernel(const float* __restrict__ wSb,
                                    const float* __restrict__ Ebuf,
                                    float* __restrict__ Ibuf) {
  int wave = (blockIdx.x * blockDim.x + threadIdx.x) >> 5;  // b*H
  int lane = threadIdx.x & 31;
  int h = wave & (Hdim - 1);
  int b = wave >> 6;
  float2 wS = ((const float2*)wSb)[h * NST + lane];
  float Ir = 0.f, Ii = 0.f;
  const float2* E = ((const float2*)Ebuf) + ((long)h * NSQ + b * NSEG) * NST + lane;
  float2*       I = ((float2*)Ibuf)       + ((long)h * NSQ + b * NSEG) * NST + lane;
  for (int c = 0; c < NSEG; ++c) {
    I[(long)c * NST] = make_float2(Ir, Ii);
    float2 e = E[(long)c * NST];
    float nr = fmaf(wS.x, Ir, fmaf(-wS.y, Ii, e.x));
    float ni = fmaf(wS.x, Ii, fmaf(wS.y, Ir, e.y));
    Ir = nr; Ii = ni;
  }
}

// ---------------------------------------------------------------------------
// Kernel F: cross-segment correction via WMMA (Phi0 @ Lam^c I), fused with
// FiLM + ReLU + residual, writing out[b,l,h].
// ---------------------------------------------------------------------------
__global__ void fixup_film_kernel(const float* __restrict__ yloc,
                                  const float* __restrict__ x,
                                  const float* __restrict__ Phi,
                                  const float* __restrict__ lam2,
                                  const float* __restrict__ Ibuf,
                                  const float* __restrict__ gb,
                                  const float* __restrict__ res_w,
                                  float* __restrict__ out) {
  int wave = (blockIdx.x * blockDim.x + threadIdx.x) >> 5;  // h*32 + tile
  int lane = threadIdx.x & 31;
  int tile = wave & 31;
  int h    = wave >> 5;
  int lid  = lane & 15;
  int bit  = lane >> 4;

  int seq  = tile * 16 + lid;
  int bcol = seq >> 6;
  int segc = seq & (NSEG - 1);
  const float* colbase = yloc + ((long)(bcol * Hdim + h)) * Ldim + segc * SEG;
  long row0 = (long)bcol * Ldim + segc * SEG;           // token row in (B*L)
  const float* xb = x + row0 * Hdim + h;
  float*       ob = out + row0 * Hdim + h;

  v2f ph[16];
  const float* phirow = Phi + ((long)h * Tc + lid) * 64 + 2 * bit;
#pragma unroll
  for (int ks = 0; ks < 16; ++ks) ph[ks] = *(const v2f*)(phirow + 4 * ks);
  float2 lm[16];
  const float2* lamh = ((const float2*)lam2) + h * NST + bit;
#pragma unroll
  for (int m = 0; m < 16; ++m) lm[m] = lamh[2 * m];

  v2f SB[16];
  const float2* ib = ((const float2*)Ibuf) + ((long)h * NSQ + seq) * NST;
#pragma unroll
  for (int m = 0; m < 16; ++m) {
    float2 v = ib[2 * m + bit];
    v2f sv = {v.x, v.y};
    SB[m] = sv;
  }

  float g  = gb[bcol * 2 * Hdim + h];
  float bb = gb[bcol * 2 * Hdim + Hdim + h];
  float rw = res_w[h];

  for (int c = 0; c < CHPSEG; ++c) {
    v8f acc = {0.f, 0.f, 0.f, 0.f, 0.f, 0.f, 0.f, 0.f};
#pragma unroll
    for (int ks = 0; ks < 16; ++ks)
      acc = __builtin_amdgcn_wmma_f32_16x16x4_f32(
          false, ph[ks], false, SB[ks], (short)0, acc, false, false);
    // advance S by w^16 for next chunk
#pragma unroll
    for (int m = 0; m < 16; ++m) {
      float sr = SB[m][0], si = SB[m][1];
      float nr = lm[m].x * sr - lm[m].y * si;
      float ni = lm[m].x * si + lm[m].y * sr;
      SB[m][0] = nr; SB[m][1] = ni;
    }
    const float* yb = colbase + c * Tc + 8 * bit;
    long tbase = (long)(c * Tc + 8 * bit);
#pragma unroll
    for (int r = 0; r < 8; ++r) {
      float yv = yb[r] + acc[r];
      float o = fmaf(g, yv, bb);
      o = o > 0.f ? o : 0.f;
      o = fmaf(xb[(tbase + r) * Hdim], rw, o);
      ob[(tbase + r) * Hdim] = o;
    }
  }
}

// ---------------------------------------------------------------------------
extern "C" void kernel_launch(void* const* d_in, const int* in_sizes, int n_in,
                              void* d_out, int out_size, void* d_ws, size_t ws_size,
                              hipStream_t stream) {
  const float* x          = (const float*)d_in[0];
  const float* cond       = (const float*)d_in[1];
  const float* W_lin      = (const float*)d_in[2];
  const float* b_lin      = (const float*)d_in[3];
  const float* log_dt     = (const float*)d_in[4];
  const float* log_A_real = (const float*)d_in[5];
  const float* A_imag     = (const float*)d_in[6];
  const float* C_ri       = (const float*)d_in[7];
  const float* D          = (const float*)d_in[8];
  const float* W_film     = (const float*)d_in[9];
  const float* b_film     = (const float*)d_in[10];
  const float* res_w      = (const float*)d_in[11];
  float* out = (float*)d_out;

  char* ws = (char*)d_ws;
  float* uy   = (float*)ws;  ws += (size_t)Bdim * Hdim * Ldim * sizeof(float);
  float* wc2  = (float*)ws;  ws += (size_t)Hdim * NST * 4 * sizeof(float);
  float* gb   = (float*)ws;  ws += (size_t)Bdim * 2 * Hdim * sizeof(float);
  float* kh   = (float*)ws;  ws += (size_t)Hdim * Tc * sizeof(float);
  float* Phi  = (float*)ws;  ws += (size_t)Hdim * Tc * 64 * sizeof(float);
  float* Psi  = (float*)ws;  ws += (size_t)Hdim * 64 * Tc * sizeof(float);
  float* lam2 = (float*)ws;  ws += (size_t)Hdim * NST * 2 * sizeof(float);
  float* wSb  = (float*)ws;  ws += (size_t)Hdim * NST * 2 * sizeof(float);
  float* Ebuf = (float*)ws;  ws += (size_t)Hdim * NSQ * NST * 2 * sizeof(float);
  float* Ibuf = (float*)ws;

  precompute_kernel<<<8, 256, 0, stream>>>(log_dt, log_A_real, A_imag, C_ri,
                                           cond, W_film, b_film, wc2, gb);
  precompute2_kernel<<<128, 256, 0, stream>>>(log_dt, log_A_real, A_imag, D,
                                              wc2, Phi, Psi, lam2, wSb, kh);
  gemm_relu_tr_kernel<<<4096, 256, 0, stream>>>(x, W_lin, b_lin, uy);
  conv_scan_kernel<<<256, 256, 0, stream>>>(uy, kh, Phi, Psi, lam2, Ebuf);
  scan_combine_kernel<<<64, 256, 0, stream>>>(wSb, Ebuf, Ibuf);
  fixup_film_kernel<<<256, 256, 0, stream>>>(uy, x, Phi, lam2, Ibuf,
                                             gb, res_w, out);
}